// MalwareGNN_26603027431731
// MI455X (gfx1250) — compile-verified
//
#include <hip/hip_runtime.h>
#include <math.h>

#define NNODES 100000
#define NEDGES 1600000
#define FEAT 128
#define HID 128
#define NCLASSES 18
#define NCENT 2
#define NGRAPHS 512

typedef float v2f __attribute__((ext_vector_type(2)));
typedef float v8f __attribute__((ext_vector_type(8)));

// ---------------------------------------------------------------------------
// GEMM: H[N x 128] = X[N x 128] @ W[128 x 128] using V_WMMA_F32_16X16X4_F32.
// Block = 256 threads = 8 waves. Block owns a 16-row slab; wave w owns the
// 16-col tile at col0 = w*16. K=128 -> 32 unrolled WMMAs per wave.
//
// f32 16x4 A fragment layout (per ISA 7.12.2): lane L holds M = L&15,
// VGPR0/1 = K = khalf / khalf+1 where khalf = (L>>4)*2. B (4x16) mirrors it
// with N = L&15. C/D: VGPR i holds row (i + (L>=16 ? 8 : 0)), col = L&15.
// ---------------------------------------------------------------------------
__global__ __launch_bounds__(256) void gemm128_wmma_kernel(
    const float* __restrict__ X, const float* __restrict__ W,
    float* __restrict__ H) {
  const int lane  = threadIdx.x & 31;
  const int wave  = threadIdx.x >> 5;
  const int row0  = blockIdx.x << 4;
  const int col0  = wave << 4;
  const int mn    = lane & 15;
  const int khalf = (lane >> 4) << 1;

  const float* __restrict__ Xr = X + (size_t)(row0 + mn) * FEAT;
  const float* __restrict__ Wc = W + col0 + mn;

  v8f acc = {};
#pragma unroll
  for (int k = 0; k < FEAT; k += 4) {
    v2f a, b;
    a.x = Xr[k + khalf];
    a.y = Xr[k + khalf + 1];
    b.x = Wc[(size_t)(k + khalf) * HID];
    b.y = Wc[(size_t)(k + khalf + 1) * HID];
    acc = __builtin_amdgcn_wmma_f32_16x16x4_f32(
        /*neg_a=*/false, a, /*neg_b=*/false, b,
        /*c_mod=*/(short)0, acc, /*reuse_a=*/false, /*reuse_b=*/false);
  }

  const int r0 = row0 + ((lane >> 4) << 3);
  float* __restrict__ Hp = H + (size_t)r0 * HID + col0 + mn;
#pragma unroll
  for (int i = 0; i < 8; ++i) Hp[(size_t)i * HID] = acc[i];
}

// ---------------------------------------------------------------------------
__global__ __launch_bounds__(256) void fill_kernel(float* __restrict__ p,
                                                   float v, int n) {
  int i = blockIdx.x * 256 + threadIdx.x;
  if (i < n) p[i] = v;
}

__global__ __launch_bounds__(256) void degree_kernel(
    const int* __restrict__ dst, float* __restrict__ deg) {
  int e = blockIdx.x * 256 + threadIdx.x;  // grid sized exactly to NEDGES
  unsafeAtomicAdd(&deg[dst[e]], 1.0f);
}

__global__ __launch_bounds__(256) void rsqrt_kernel(float* __restrict__ p,
                                                    int n) {
  int i = blockIdx.x * 256 + threadIdx.x;
  if (i < n) p[i] = rsqrtf(p[i]);  // deg -> deg^-1/2 in place
}

// One wave per edge: lanes 0..31 each move a float4 (128 feats total).
// src/dst/norm are wave-uniform -> scalar loads; atomics hit L2-resident agg.
__global__ __launch_bounds__(256) void scatter_kernel(
    const float* __restrict__ H, const int* __restrict__ src,
    const int* __restrict__ dst, const float* __restrict__ dinv,
    float* __restrict__ agg) {
  const unsigned tid = blockIdx.x * 256u + threadIdx.x;
  const unsigned e  = tid >> 5;
  const unsigned f4 = (tid & 31u) << 2;
  const int s = src[e];
  const int d = dst[e];
  const float norm = dinv[s] * dinv[d];
  const float4 v = *(const float4*)(H + (size_t)s * HID + f4);
  float* ap = agg + (size_t)d * HID + f4;
  unsafeAtomicAdd(ap + 0, v.x * norm);
  unsafeAtomicAdd(ap + 1, v.y * norm);
  unsafeAtomicAdd(ap + 2, v.z * norm);
  unsafeAtomicAdd(ap + 3, v.w * norm);
}

// agg[n] = agg[n] + H[n] * deg^-1 + bias ; optional ReLU. In-place on agg so
// it becomes the next layer's input.
__global__ __launch_bounds__(256) void finalize_kernel(
    float* __restrict__ agg, const float* __restrict__ H,
    const float* __restrict__ dinv, const float* __restrict__ bias, int relu) {
  const unsigned tid = blockIdx.x * 256u + threadIdx.x;
  const unsigned n  = tid >> 5;
  const unsigned f4 = (tid & 31u) << 2;
  const float di = dinv[n];
  const float invdeg = di * di;
  float4 a = *(float4*)(agg + (size_t)n * HID + f4);
  const float4 h = *(const float4*)(H + (size_t)n * HID + f4);
  a.x += h.x * invdeg + bias[f4 + 0];
  a.y += h.y * invdeg + bias[f4 + 1];
  a.z += h.z * invdeg + bias[f4 + 2];
  a.w += h.w * invdeg + bias[f4 + 3];
  if (relu) {
    a.x = fmaxf(a.x, 0.0f); a.y = fmaxf(a.y, 0.0f);
    a.z = fmaxf(a.z, 0.0f); a.w = fmaxf(a.w, 0.0f);
  }
  *(float4*)(agg + (size_t)n * HID + f4) = a;
}

__global__ __launch_bounds__(256) void pool_sum_kernel(
    const float* __restrict__ h, const int* __restrict__ batch,
    float* __restrict__ psum) {
  const unsigned tid = blockIdx.x * 256u + threadIdx.x;
  const unsigned n  = tid >> 5;
  const unsigned f4 = (tid & 31u) << 2;
  const int b = batch[n];
  const float4 v = *(const float4*)(h + (size_t)n * HID + f4);
  float* sp = psum + (size_t)b * HID + f4;
  unsafeAtomicAdd(sp + 0, v.x);
  unsafeAtomicAdd(sp + 1, v.y);
  unsafeAtomicAdd(sp + 2, v.z);
  unsafeAtomicAdd(sp + 3, v.w);
}

__global__ __launch_bounds__(256) void pool_cnt_kernel(
    const int* __restrict__ batch, float* __restrict__ pcnt, int n) {
  int i = blockIdx.x * 256 + threadIdx.x;
  if (i < n) unsafeAtomicAdd(&pcnt[batch[i]], 1.0f);
}

// One 64-thread block per graph. g = mean(h), dist to 18x2 centroids, min
// over the 2 centroids, y = -dist; soft-accept head (RUNNING_MEAN=1,
// RUNNING_VAR=0, AC_STD_LIM=5).
__global__ __launch_bounds__(64) void centroid_kernel(
    const float* __restrict__ psum, const float* __restrict__ pcnt,
    const float* __restrict__ centroids, const float* __restrict__ std_scale,
    const float* __restrict__ ac_temp, float* __restrict__ out) {
  __shared__ float g[HID];
  __shared__ float dpair[NCLASSES * NCENT];
  __shared__ float dmin[NCLASSES];
  const int b = blockIdx.x;
  const int t = threadIdx.x;
  const float invc = 1.0f / fmaxf(pcnt[b], 1.0f);
  g[t]      = psum[(size_t)b * HID + t] * invc;
  g[t + 64] = psum[(size_t)b * HID + t + 64] * invc;
  __syncthreads();
  if (t < NCLASSES * NCENT) {
    const float* c = centroids + (size_t)t * HID;
    float acc = 0.0f;
    for (int k = 0; k < HID; ++k) {
      float d = c[k] - g[k];
      acc += d * d;
    }
    dpair[t] = sqrtf(acc);
  }
  __syncthreads();
  if (t < NCLASSES) {
    float d = fminf(dpair[2 * t], dpair[2 * t + 1]);
    dmin[t] = d;
    out[(size_t)b * NCLASSES + t] = -d;
  }
  __syncthreads();
  if (t == 0) {
    float mind = dmin[0];
    for (int c = 1; c < NCLASSES; ++c) mind = fminf(mind, dmin[c]);
    float ss = fminf(fmaxf(std_scale[0], 0.0f), 5.0f);
    float max_ac = 1.0f + ss * sqrtf(0.0f);  // RUNNING_MEAN + clip*sqrt(VAR)
    float accept = max_ac - mind;
    float soft = 1.0f / (1.0f + expf(-accept / ac_temp[0]));
    out[(size_t)NGRAPHS * NCLASSES + b] = soft;
  }
}

// ---------------------------------------------------------------------------
extern "C" void kernel_launch(void* const* d_in, const int* in_sizes, int n_in,
                              void* d_out, int out_size, void* d_ws,
                              size_t ws_size, hipStream_t stream) {
  const float* x     = (const float*)d_in[0];
  const int*   ei    = (const int*)d_in[1];
  const int*   batch = (const int*)d_in[2];
  const float* W1 = (const float*)d_in[3];
  const float* b1 = (const float*)d_in[4];
  const float* W2 = (const float*)d_in[5];
  const float* b2 = (const float*)d_in[6];
  const float* W3 = (const float*)d_in[7];
  const float* b3 = (const float*)d_in[8];
  const float* cent      = (const float*)d_in[9];
  const float* std_scale = (const float*)d_in[10];
  const float* ac_temp   = (const float*)d_in[11];
  const int* src = ei;
  const int* dst = ei + NEDGES;

  const size_t NH = (size_t)NNODES * HID;  // 12.8M floats
  float* ws   = (float*)d_ws;
  float* Hbuf = ws;              // GEMM output
  float* agg  = ws + NH;         // scatter accumulator / layer output
  float* dinv = ws + 2 * NH;     // NNODES
  float* psum = dinv + NNODES;   // NGRAPHS*HID
  float* pcnt = psum + (size_t)NGRAPHS * HID;  // NGRAPHS

  const int gN16   = NNODES / 16;                 // 6250 (exact)
  const int gEdge  = NEDGES / 256;                // 6250 (exact)
  const int gElem  = (NNODES * 32) / 256;         // 12500 (exact)
  const int gScat  = (int)(((size_t)NEDGES * 32) / 256);  // 200000 (exact)
  const int gZero  = (int)((NH + 255) / 256);
  const int gNode  = (NNODES + 255) / 256;
  const int poolSz = NGRAPHS * HID + NGRAPHS;
  const int gPoolZ = (poolSz + 255) / 256;

  // degrees (with self-loop) -> dinv = deg^-1/2
  fill_kernel<<<gNode, 256, 0, stream>>>(dinv, 1.0f, NNODES);
  degree_kernel<<<gEdge, 256, 0, stream>>>(dst, dinv);
  rsqrt_kernel<<<gNode, 256, 0, stream>>>(dinv, NNODES);

  const float* layer_in = x;
  const float* Ws[3] = {W1, W2, W3};
  const float* bs[3] = {b1, b2, b3};
  for (int l = 0; l < 3; ++l) {
    gemm128_wmma_kernel<<<gN16, 256, 0, stream>>>(layer_in, Ws[l], Hbuf);
    fill_kernel<<<gZero, 256, 0, stream>>>(agg, 0.0f, (int)NH);
    scatter_kernel<<<gScat, 256, 0, stream>>>(Hbuf, src, dst, dinv, agg);
    finalize_kernel<<<gElem, 256, 0, stream>>>(agg, Hbuf, dinv, bs[l],
                                               l < 2 ? 1 : 0);
    layer_in = agg;
  }

  // global mean pool + centroid head
  fill_kernel<<<gPoolZ, 256, 0, stream>>>(psum, 0.0f, poolSz);
  pool_sum_kernel<<<gElem, 256, 0, stream>>>(agg, batch, psum);
  pool_cnt_kernel<<<gNode, 256, 0, stream>>>(batch, pcnt, NNODES);
  centroid_kernel<<<NGRAPHS, 64, 0, stream>>>(psum, pcnt, cent, std_scale,
                                              ac_temp, (float*)d_out);
}